// GNNBlock_80032420594207
// MI455X (gfx1250) — compile-verified
//
#include <hip/hip_runtime.h>

typedef float v2f __attribute__((ext_vector_type(2)));
typedef float v8f __attribute__((ext_vector_type(8)));

#define D_CH 128
#define EPS_LN 1e-5f

// ---------------------------------------------------------------- zero scratch
__global__ void zero_kernel(float* __restrict__ p, long long n) {
    long long stride = (long long)gridDim.x * blockDim.x;
    for (long long i = (long long)blockIdx.x * blockDim.x + threadIdx.x; i < n; i += stride)
        p[i] = 0.0f;
}

// ------------------------------------------------- global sum / sumsq (float64)
__global__ void reduce_kernel(const float* __restrict__ x, long long n,
                              double* __restrict__ acc) {
    __shared__ double ss[256];
    __shared__ double sq[256];
    double s = 0.0, q = 0.0;
    long long stride = (long long)gridDim.x * blockDim.x;
    for (long long i = (long long)blockIdx.x * blockDim.x + threadIdx.x; i < n; i += stride) {
        double v = (double)x[i];
        s += v;
        q += v * v;
    }
    ss[threadIdx.x] = s;
    sq[threadIdx.x] = q;
    __syncthreads();
    for (int o = 128; o > 0; o >>= 1) {
        if ((int)threadIdx.x < o) {
            ss[threadIdx.x] += ss[threadIdx.x + o];
            sq[threadIdx.x] += sq[threadIdx.x + o];
        }
        __syncthreads();
    }
    if (threadIdx.x == 0) {
        atomicAdd(&acc[0], ss[0]);
        atomicAdd(&acc[1], sq[0]);
    }
}

__global__ void stats_kernel(const double* __restrict__ acc, float* __restrict__ ms,
                             long long n) {
    double mean = acc[0] / (double)n;
    double var  = acc[1] / (double)n - mean * mean;
    if (var < 0.0) var = 0.0;
    double stdv = sqrt(var);
    ms[0] = (float)mean;
    ms[1] = (float)(1.0 / (stdv + (double)EPS_LN));
}

// --------------------------- h = relu(((x-mean)*scale)*ln_w + ln_b) * dropout
__global__ void h_kernel(const float4* __restrict__ x, const float4* __restrict__ mask,
                         const float* __restrict__ lnw, const float* __restrict__ lnb,
                         const float* __restrict__ ms, float4* __restrict__ hout, int nvec) {
    int i = blockIdx.x * blockDim.x + threadIdx.x;
    if (i >= nvec) return;
    float mean = ms[0], scale = ms[1];
    int d = (i & 31) * 4;  // channel base within the 128-wide row
    float4 xv = x[i];
    float4 mv = mask[i];
    float4 r;
    r.x = fmaxf((xv.x - mean) * scale * lnw[d + 0] + lnb[d + 0], 0.0f) * mv.x;
    r.y = fmaxf((xv.y - mean) * scale * lnw[d + 1] + lnb[d + 1], 0.0f) * mv.y;
    r.z = fmaxf((xv.z - mean) * scale * lnw[d + 2] + lnb[d + 2], 0.0f) * mv.z;
    r.w = fmaxf((xv.w - mean) * scale * lnw[d + 3] + lnb[d + 3], 0.0f) * mv.w;
    hout[i] = r;
}

// ----------------------------- one wave per edge: 32 lanes x float4 = 128 chans
__global__ void scatter_kernel(const int* __restrict__ ei, int E,
                               const float4* __restrict__ h,
                               float* __restrict__ agg, float* __restrict__ deg) {
    int t = blockIdx.x * blockDim.x + threadIdx.x;
    int e = t >> 5;
    int lane = t & 31;
    if (e >= E) return;
    int src = ei[e];
    int dst = ei[E + e];
    float4 v = h[(size_t)src * 32 + lane];
    float* a = agg + (size_t)dst * D_CH + lane * 4;
    atomicAdd(a + 0, v.x);
    atomicAdd(a + 1, v.y);
    atomicAdd(a + 2, v.z);
    atomicAdd(a + 3, v.w);
    if (lane == 0) atomicAdd(&deg[dst], 1.0f);
}

// ---------- out[16x128 tile] = [agg*rdeg | h] (16x256) @ [w_l ; w_r] (256x128) + b
// one wave per 16x16 output tile, f32 WMMA 16x16x4, K-loop of 64 steps
__global__ void __launch_bounds__(256)
gemm_kernel(const float* __restrict__ h, const float* __restrict__ agg,
            const float* __restrict__ deg, const float* __restrict__ wl,
            const float* __restrict__ wr, const float* __restrict__ bl,
            float* __restrict__ out) {
    const int wave = threadIdx.x >> 5;          // 0..7 -> column tile
    const int lane = threadIdx.x & 31;
    const int r = lane & 15;                    // row within tile (A), col within tile (B/D)
    const int khalf = (lane >> 4) * 2;          // lanes 16-31 hold K+2..K+3
    const int row_base = blockIdx.x * 16;
    const int arow = row_base + r;
    const int col = wave * 16 + r;

    const float dg = deg[arow];
    const float rdeg = 1.0f / fmaxf(dg, 1.0f);
    const float* __restrict__ arow_p = agg + (size_t)arow * D_CH;
    const float* __restrict__ hrow_p = h + (size_t)arow * D_CH;

    v8f acc = {};

    // phase 1: (agg * rdeg) @ w_l   (K = 0..127)
    #pragma unroll 4
    for (int k = 0; k < D_CH; k += 4) {
        int ka = k + khalf;
        v2f a, b;
        a.x = arow_p[ka + 0] * rdeg;
        a.y = arow_p[ka + 1] * rdeg;
        b.x = wl[(ka + 0) * D_CH + col];
        b.y = wl[(ka + 1) * D_CH + col];
        acc = __builtin_amdgcn_wmma_f32_16x16x4_f32(false, a, false, b,
                                                    (short)0, acc, false, false);
    }
    // phase 2: h @ w_r              (K = 128..255)
    #pragma unroll 4
    for (int k = 0; k < D_CH; k += 4) {
        int ka = k + khalf;
        v2f a, b;
        a.x = hrow_p[ka + 0];
        a.y = hrow_p[ka + 1];
        b.x = wr[(ka + 0) * D_CH + col];
        b.y = wr[(ka + 1) * D_CH + col];
        acc = __builtin_amdgcn_wmma_f32_16x16x4_f32(false, a, false, b,
                                                    (short)0, acc, false, false);
    }

    // D layout: VGPR i -> M = i (lanes 0-15) / i+8 (lanes 16-31), N = lane&15
    const float bias = bl[col];
    const int out_row0 = row_base + (lane >> 4) * 8;
    #pragma unroll
    for (int i = 0; i < 8; ++i)
        out[(size_t)(out_row0 + i) * D_CH + col] = acc[i] + bias;
}

// ---------------------------------------------------------------------- launch
extern "C" void kernel_launch(void* const* d_in, const int* in_sizes, int n_in,
                              void* d_out, int out_size, void* d_ws, size_t ws_size,
                              hipStream_t stream) {
    const float* x    = (const float*)d_in[0];
    const int*   ei   = (const int*)d_in[1];
    const float* mask = (const float*)d_in[2];
    const float* lnw  = (const float*)d_in[3];
    const float* lnb  = (const float*)d_in[4];
    const float* wl   = (const float*)d_in[5];
    const float* bl   = (const float*)d_in[6];
    const float* wr   = (const float*)d_in[7];
    float* out = (float*)d_out;

    const int n = in_sizes[0] / D_CH;      // 50000 nodes
    const int E = in_sizes[1] / 2;         // 800000 edges
    const long long H = (long long)n * D_CH;

    // workspace layout: h | agg | deg | {double sum, double sumsq, float mean, float scale}
    float* ws   = (float*)d_ws;
    float* h    = ws;
    float* agg  = ws + H;
    float* deg  = ws + 2 * H;
    long long so = (2 * H + n + 3) & ~3LL;       // 16B-aligned scalar area
    double* acc = (double*)(ws + so);
    float*  ms  = (float*)(acc + 2);

    // 1) zero agg + deg + scalar accumulators
    long long zn = (so - H) + 6;
    zero_kernel<<<2048, 256, 0, stream>>>(agg, zn);

    // 2) global mean / sumsq reduction in f64
    reduce_kernel<<<2048, 256, 0, stream>>>(x, H, acc);

    // 3) mean and 1/(std+eps)
    stats_kernel<<<1, 1, 0, stream>>>(acc, ms, H);

    // 4) fused layernorm + relu + dropout -> h
    int nvec = (int)(H / 4);
    h_kernel<<<(nvec + 255) / 256, 256, 0, stream>>>(
        (const float4*)x, (const float4*)mask, lnw, lnb, ms, (float4*)h, nvec);

    // 5) edge-parallel scatter-add (one wave32 per edge, L2-resident atomics)
    long long sthreads = (long long)E * 32;
    scatter_kernel<<<(int)((sthreads + 255) / 256), 256, 0, stream>>>(
        ei, E, (const float4*)h, agg, deg);

    // 6) fused mean-normalize + dual GEMM + bias via f32 WMMA
    gemm_kernel<<<n / 16, 256, 0, stream>>>(h, agg, deg, wl, wr, bl, out);
}